// LeftDRegularGraph_54571854463052
// MI455X (gfx1250) — compile-verified
//
#include <hip/hip_runtime.h>

// LeftDRegularGraph: gumbel-softmax top-k straight-through, b=16, m=512, n=4096, d=16.
// Single data pass: online softmax + top-16 (value,index) selection network, then
// zero-fill + 16-element scatter. Traffic = 128MB noise (NT, async->LDS staged) +
// 8MB param + 128MB out = 264MB => ~11.3us floor @ 23.3 TB/s. No GEMM -> no WMMA;
// the CDNA5-specific path exercised is the async global->LDS engine (ASYNCcnt).

#define B_  16
#define M_  512
#define N_  4096
#define D_  16
#define LG2N 12

#define CH  32                       // rows per staged chunk
#define NCHUNK (M_ / CH)             // 16 chunks per column
#define WAVES 8                      // waves per block (wave32)
#define COLS_PER_WAVE 32             // one column per lane
#define BUF_FLOATS (CH * COLS_PER_WAVE)   // 1024 floats = 4KB per buffer

// total static LDS: 2 buffers * 8 waves * 4KB = 64KB
__global__ __launch_bounds__(256) void lrg_topk_softmax_kernel(
    const float* __restrict__ param,   // (1, M, N)
    const float* __restrict__ noise,   // (B, M, N) uniform(0,1)
    const float* __restrict__ scalar,  // (1,)
    float* __restrict__ out)           // (B, M, N)
{
    __shared__ float lds[2 * WAVES * BUF_FLOATS];

    const int tid  = threadIdx.x;
    const int wave = tid >> 5;
    const int lane = tid & 31;

    // This wave's 32 consecutive columns (all inside one batch b since 256 | N)
    const int cbase = blockIdx.x * (WAVES * COLS_PER_WAVE) + wave * COLS_PER_WAVE;
    const int b     = cbase >> LG2N;
    const int n0    = cbase & (N_ - 1);
    const int ncol  = n0 + lane;

    // async b128 pattern: each instruction moves 4 rows x 32 cols (512B).
    // lane l -> row (l>>3), cols ((l&7)*4 .. +3)
    const unsigned lanehi = (unsigned)(lane >> 3);
    const unsigned laneq  = (unsigned)((lane & 7) << 2);

    const unsigned g_wave_byte = (unsigned)(((size_t)b * M_ * N_ + (size_t)n0) * 4u);
    const unsigned g_lane_byte = (lanehi * N_ + laneq) * 4u;

    const unsigned lds_base    = (unsigned)(size_t)(&lds[0]); // low 32 bits = LDS offset
    const unsigned lds_wave    = lds_base + (unsigned)(wave * 2 * BUF_FLOATS) * 4u;
    const unsigned l_lane_byte = (lanehi * COLS_PER_WAVE + laneq) * 4u;

    const float* bufw = &lds[wave * 2 * BUF_FLOATS];

    // Stage one 32x32 chunk of noise into LDS buffer bufsel via the async DMA path.
    // Noise is strictly single-use: non-temporal so it never evicts L2-resident param.
    auto issue_chunk = [&](int m0, int bufsel) {
        unsigned vaddr = g_wave_byte + (unsigned)(m0 * N_) * 4u + g_lane_byte;
        unsigned vlds  = lds_wave + (unsigned)(bufsel * BUF_FLOATS) * 4u + l_lane_byte;
        #pragma unroll
        for (int i = 0; i < CH / 4; ++i) {
            asm volatile("global_load_async_to_lds_b128 %0, %1, %2 th:TH_LOAD_NT"
                         :: "v"(vlds), "v"(vaddr), "s"(noise) : "memory");
            vaddr += 4u * N_ * 4u;             // next 4 rows in global
            vlds  += 4u * COLS_PER_WAVE * 4u;  // next 4 rows in LDS
        }
    };

    const float NEG_INF = -__builtin_inff();

    // ---------------- Pass 1: online softmax stats + top-16 (value,index) ----------------
    float mx   = NEG_INF;
    float ssum = 0.0f;
    float t[D_];                       // sorted ascending; t[0] = current 16th-largest
    int   ix[D_];                      // matching m-indices
    #pragma unroll
    for (int i = 0; i < D_; ++i) { t[i] = NEG_INF; ix[i] = 0; }

    issue_chunk(0, 0);
    for (int ch = 0; ch < NCHUNK; ++ch) {
        if (ch + 1 < NCHUNK) {
            issue_chunk((ch + 1) * CH, (ch + 1) & 1);
            asm volatile("s_wait_asynccnt 8" ::: "memory");  // chunk ch complete (in-order)
        } else {
            asm volatile("s_wait_asynccnt 0" ::: "memory");
        }
        const float* bp = bufw + (ch & 1) * BUF_FLOATS;
        const int mbase = ch * CH;
        #pragma unroll 4
        for (int r = 0; r < CH; ++r) {
            float u  = bp[r * COLS_PER_WAVE + lane];
            float pm = param[(size_t)(mbase + r) * N_ + ncol];
            float g  = -__logf(-__logf(u));
            float lg = fmaf(g, 0.001f, pm);
            // online (flash) softmax
            if (lg > mx) { ssum *= __expf(mx - lg); mx = lg; }
            ssum += __expf(lg - mx);
            // sorted insert-evict-min network carrying indices; static register
            // indices only (ascending in-place => old i+1 entries still live)
            if (lg > t[0]) {
                const int mrow = mbase + r;
                #pragma unroll
                for (int i = 0; i < D_ - 1; ++i) {
                    float tn = t[i + 1];               // old neighbor value
                    int   in = ix[i + 1];              // old neighbor index
                    bool  gt_i = lg > t[i];
                    bool  le_n = !(lg > tn);
                    ix[i] = gt_i ? (le_n ? mrow : in) : ix[i];
                    t[i]  = fmaxf(t[i], fminf(tn, lg));
                }
                bool gt_l = lg > t[D_ - 1];
                ix[D_ - 1] = gt_l ? mrow : ix[D_ - 1];
                t[D_ - 1]  = fmaxf(t[D_ - 1], lg);
            }
        }
    }

    // ---------------- Pass 2: zero-fill + 16-element scatter (no global reads) ----------
    const float inv_s = 1.0f / ssum;
    const float scl   = fmaxf(0.01f, scalar[0]);

    float vals[D_];
    #pragma unroll
    for (int i = 0; i < D_; ++i) {
        float p = __expf(t[i] - mx) * inv_s;          // softmax prob of a selected row
        // phi = y_hard - sg(p) + p : (1-p)+p on-support, exactly +0 off-support
        vals[i] = ((1.0f - p) + p) * 0.25f * scl;     // /sqrt(16) then *scale
    }

    float* outc = out + (size_t)b * M_ * N_ + (size_t)ncol;

    // coalesced zero-fill of this lane's column (128B/row across the wave)
    #pragma unroll 8
    for (int m = 0; m < M_; ++m)
        outc[(size_t)m * N_] = 0.0f;

    // overwrite the 16 selected rows; same-wave stores complete in order (ISA 7.3)
    #pragma unroll
    for (int i = 0; i < D_; ++i)
        outc[(size_t)ix[i] * N_] = vals[i];
}

extern "C" void kernel_launch(void* const* d_in, const int* in_sizes, int n_in,
                              void* d_out, int out_size, void* d_ws, size_t ws_size,
                              hipStream_t stream) {
    const float* param  = (const float*)d_in[0];  // (1, 512, 4096) f32
    const float* noise  = (const float*)d_in[1];  // (16, 512, 4096) f32
    const float* scalar = (const float*)d_in[2];  // (1,) f32
    float* out = (float*)d_out;                   // (16, 512, 4096) f32

    const int total_cols = B_ * N_;                               // 65536
    const int blocks = total_cols / (WAVES * COLS_PER_WAVE);      // 256
    lrg_topk_softmax_kernel<<<dim3(blocks), dim3(256), 0, stream>>>(param, noise, scalar, out);
}